// GaussianSplat_21045339751012
// MI455X (gfx1250) — compile-verified
//
#include <hip/hip_runtime.h>
#include <stdint.h>

#define NEAR_PLANE 0.01f
#define BLUR 0.3f
#define MAX_ALPHA 0.999f

#define NG 1024
#define NCAM 2
#define W_IMG 128
#define H_IMG 128
#define NPIX (W_IMG * H_IMG)
#define FSTRIDE (NCAM * NG) /* per-field stride in workspace */
#define CHUNK 128
#define NCHUNK (NG / CHUNK)

typedef float v2f __attribute__((ext_vector_type(2)));
typedef float v8f __attribute__((ext_vector_type(8)));

__device__ __forceinline__ float sigmoidf_(float x) { return 1.f / (1.f + __expf(-x)); }

// ---------------------------------------------------------------------------
// Phase 1: per-gaussian projection.  U holds 10 unsorted fields:
//   0:mean2d.x 1:mean2d.y 2:ea 3:eb 4:ec 5:opac(0 if invalid) 6:r 7:g 8:b 9:tz
// where (ea,eb,ec) are the conic coefficients pre-scaled by -0.5*log2(e) so the
// render loop computes alpha = op * exp2(ea*dx^2 + eb*dx*dy + ec*dy^2).
// indexed U[f*FSTRIDE + cam*NG + n]
// ---------------------------------------------------------------------------
__global__ __launch_bounds__(256) void prep_kernel(
    const float* __restrict__ vm, const float* __restrict__ Ks,
    const float* __restrict__ means, const float* __restrict__ quats,
    const float* __restrict__ logsc, const float* __restrict__ opl,
    const float* __restrict__ cl, float* __restrict__ U) {
  int id = blockIdx.x * blockDim.x + threadIdx.x;
  if (id >= NCAM * NG) return;
  const int cam = id >> 10, n = id & (NG - 1);

  // normalized quaternion -> rotation
  float qw = quats[n * 4 + 0], qx = quats[n * 4 + 1];
  float qy = quats[n * 4 + 2], qz = quats[n * 4 + 3];
  float qn = rsqrtf(qw * qw + qx * qx + qy * qy + qz * qz);
  qw *= qn; qx *= qn; qy *= qn; qz *= qn;
  float R00 = 1.f - 2.f * (qy * qy + qz * qz), R01 = 2.f * (qx * qy - qw * qz), R02 = 2.f * (qx * qz + qw * qy);
  float R10 = 2.f * (qx * qy + qw * qz), R11 = 1.f - 2.f * (qx * qx + qz * qz), R12 = 2.f * (qy * qz - qw * qx);
  float R20 = 2.f * (qx * qz - qw * qy), R21 = 2.f * (qy * qz + qw * qx), R22 = 1.f - 2.f * (qx * qx + qy * qy);

  float s0 = __expf(logsc[n * 3 + 0]), s1 = __expf(logsc[n * 3 + 1]), s2 = __expf(logsc[n * 3 + 2]);
  float M00 = R00 * s0, M01 = R01 * s1, M02 = R02 * s2;
  float M10 = R10 * s0, M11 = R11 * s1, M12 = R12 * s2;
  float M20 = R20 * s0, M21 = R21 * s1, M22 = R22 * s2;
  // cov3d = M M^T (symmetric)
  float C00 = M00 * M00 + M01 * M01 + M02 * M02;
  float C01 = M00 * M10 + M01 * M11 + M02 * M12;
  float C02 = M00 * M20 + M01 * M21 + M02 * M22;
  float C11 = M10 * M10 + M11 * M11 + M12 * M12;
  float C12 = M10 * M20 + M11 * M21 + M12 * M22;
  float C22 = M20 * M20 + M21 * M21 + M22 * M22;

  const float* V = vm + cam * 16;
  float W00 = V[0], W01 = V[1], W02 = V[2], t0 = V[3];
  float W10 = V[4], W11 = V[5], W12 = V[6], t1 = V[7];
  float W20 = V[8], W21 = V[9], W22 = V[10], t2 = V[11];
  float mx = means[n * 3 + 0], my = means[n * 3 + 1], mz = means[n * 3 + 2];
  float tcx = W00 * mx + W01 * my + W02 * mz + t0;
  float tcy = W10 * mx + W11 * my + W12 * mz + t1;
  float tcz = W20 * mx + W21 * my + W22 * mz + t2;

  const float* Kc = Ks + cam * 9;
  float fx = Kc[0], cx = Kc[2], fy = Kc[4], cy = Kc[5];
  float tzs = tcz > NEAR_PLANE ? tcz : NEAR_PLANE;
  float iz = 1.f / tzs;
  float m2x = fx * tcx * iz + cx, m2y = fy * tcy * iz + cy;

  // cov_cam = W * cov3d * W^T
  float B00 = W00 * C00 + W01 * C01 + W02 * C02;
  float B01 = W00 * C01 + W01 * C11 + W02 * C12;
  float B02 = W00 * C02 + W01 * C12 + W02 * C22;
  float B10 = W10 * C00 + W11 * C01 + W12 * C02;
  float B11 = W10 * C01 + W11 * C11 + W12 * C12;
  float B12 = W10 * C02 + W11 * C12 + W12 * C22;
  float B20 = W20 * C00 + W21 * C01 + W22 * C02;
  float B21 = W20 * C01 + W21 * C11 + W22 * C12;
  float B22 = W20 * C02 + W21 * C12 + W22 * C22;
  float CC00 = B00 * W00 + B01 * W01 + B02 * W02;
  float CC01 = B00 * W10 + B01 * W11 + B02 * W12;
  float CC02 = B00 * W20 + B01 * W21 + B02 * W22;
  float CC11 = B10 * W10 + B11 * W11 + B12 * W12;
  float CC12 = B10 * W20 + B11 * W21 + B12 * W22;
  float CC22 = B20 * W20 + B21 * W21 + B22 * W22;

  // J rows: (J00,0,J02), (0,J11,J12)
  float J00 = fx * iz, J02 = -fx * tcx * iz * iz;
  float J11 = fy * iz, J12 = -fy * tcy * iz * iz;
  float v0x = CC00 * J00 + CC02 * J02;
  float v0y = CC01 * J00 + CC12 * J02;
  float v0z = CC02 * J00 + CC22 * J02;
  float cov00 = J00 * v0x + J02 * v0z;
  float cov01 = J11 * v0y + J12 * v0z;
  float v1y = CC11 * J11 + CC12 * J12;
  float v1z = CC12 * J11 + CC22 * J12;
  float cov11 = J11 * v1y + J12 * v1z;

  float a = cov00 + BLUR, c2 = cov11 + BLUR, b = cov01;
  float det = a * c2 - b * b;
  bool valid = (tcz > NEAR_PLANE) && (det > 1e-12f);
  float dets = valid ? det : 1.f;
  float inv = 1.f / dets;

  const float NL2E = -0.5f * 1.44269504088896340736f; // -0.5*log2(e)
  float* Uc = U + cam * NG + n;
  Uc[0 * FSTRIDE] = m2x;
  Uc[1 * FSTRIDE] = m2y;
  Uc[2 * FSTRIDE] = NL2E * (c2 * inv);          // ea
  Uc[3 * FSTRIDE] = NL2E * (-2.f * b * inv);    // eb (covers the 2*ib term)
  Uc[4 * FSTRIDE] = NL2E * (a * inv);           // ec
  Uc[5 * FSTRIDE] = valid ? sigmoidf_(opl[n]) : 0.f;
  Uc[6 * FSTRIDE] = sigmoidf_(cl[n * 3 + 0]);
  Uc[7 * FSTRIDE] = sigmoidf_(cl[n * 3 + 1]);
  Uc[8 * FSTRIDE] = sigmoidf_(cl[n * 3 + 2]);
  Uc[9 * FSTRIDE] = tcz;
}

// ---------------------------------------------------------------------------
// Phase 2: per-camera bitonic sort of 1024 gaussians by tz, then reorder the
// 9 render fields into S[f*FSTRIDE + cam*NG + slot].
// ---------------------------------------------------------------------------
__global__ __launch_bounds__(1024) void sort_kernel(const float* __restrict__ U,
                                                    float* __restrict__ S) {
  __shared__ unsigned long long keys[NG];
  const int cam = blockIdx.x;
  const int t = threadIdx.x;

  float tz = U[9 * FSTRIDE + cam * NG + t];
  unsigned u = __float_as_uint(tz);
  u = (u & 0x80000000u) ? ~u : (u | 0x80000000u); // orderable-uint transform
  keys[t] = ((unsigned long long)u << 32) | (unsigned)t;
  __syncthreads();

  for (int k = 2; k <= NG; k <<= 1) {
    for (int j = k >> 1; j > 0; j >>= 1) {
      int ixj = t ^ j;
      if (ixj > t) {
        unsigned long long va = keys[t], vb = keys[ixj];
        bool asc = (t & k) == 0;
        if ((va > vb) == asc) { keys[t] = vb; keys[ixj] = va; }
      }
      __syncthreads();
    }
  }

  unsigned idx = (unsigned)(keys[t] & 0xffffffffu);
  const float* Ub = U + cam * NG;
  float* Sb = S + cam * NG;
#pragma unroll
  for (int f = 0; f < 9; ++f) Sb[f * FSTRIDE + t] = Ub[f * FSTRIDE + idx];
}

// ---------------------------------------------------------------------------
// Phase 3: compositing.  Block = 256 threads = 8 waves; each wave owns 16
// consecutive pixels.  Double-buffered chunks of 128 gaussians staged to LDS
// with global_load_async_to_lds_b128 (overlapped with compute, synced via
// s_wait_asynccnt + workgroup barrier).  Per 4 gaussians: lane halves compute
// alphas for (k,k+1)/(k+2,k+3) of their pixel, exchange (1-a) products via
// ds_swizzle SWAPX16, build compositing weights, and accumulate colors with
// v_wmma_f32_16x16x4_f32.
// ---------------------------------------------------------------------------
__global__ __launch_bounds__(256) void render_kernel(const float* __restrict__ S,
                                                     float* __restrict__ out) {
  __shared__ float smem[2][9 * CHUNK];
  const int cam = blockIdx.x >> 7;
  const int blk = blockIdx.x & 127;
  const int warp = threadIdx.x >> 5;
  const int lane = threadIdx.x & 31;
  const int h = lane >> 4;      // which lane-half (K offset 0 or 2)
  const int n16 = lane & 15;    // row (pixel) / column (channel) index
  const int pwave = blk * 128 + warp * 16;
  const int pix = pwave + n16;
  const float pxf = (float)(pix & (W_IMG - 1)) + 0.5f;
  const float pyf = (float)(pix >> 7) + 0.5f;
  const float* __restrict__ Sc = S + cam * NG;

  // color-column setup: lanes 0..2 fetch channel n16, others fetch ch 0 * 0
  const float cmask = (n16 < 3) ? 1.f : 0.f;
  const int ch6 = 6 + ((n16 < 3) ? n16 : 0);

  v8f acc = {0.f, 0.f, 0.f, 0.f, 0.f, 0.f, 0.f, 0.f};
  float T = 1.f;

  // stage one chunk (9 fields * 128 floats = 288 x b128) into dstbuf
  auto stage = [&](float* dstbuf, int base) {
    for (int i = threadIdx.x; i < 9 * (CHUNK / 4); i += 256) {
      const int f = i >> 5, j = (i & 31) << 2;
      const float* gp = Sc + f * FSTRIDE + base + j;
      unsigned lds_off = (unsigned)(uintptr_t)(&dstbuf[f * CHUNK + j]);
      unsigned long long ga = (unsigned long long)(uintptr_t)gp;
      asm volatile("global_load_async_to_lds_b128 %0, %1, off"
                   :: "v"(lds_off), "v"(ga) : "memory");
    }
  };

  stage(smem[0], 0);
  asm volatile("s_wait_asynccnt 0x0" ::: "memory");
  __syncthreads();

  int cur = 0;
  for (int c = 0; c < NCHUNK; ++c) {
    if (c + 1 < NCHUNK) stage(smem[cur ^ 1], (c + 1) * CHUNK);

    const float* __restrict__ bf = smem[cur];
    for (int kk = 0; kk < CHUNK; kk += 4) {
      const int g0 = kk + 2 * h; // even -> 8B-aligned pair loads
      v2f gx = *(const v2f*)&bf[0 * CHUNK + g0];
      v2f gy = *(const v2f*)&bf[1 * CHUNK + g0];
      v2f ea = *(const v2f*)&bf[2 * CHUNK + g0];
      v2f eb = *(const v2f*)&bf[3 * CHUNK + g0];
      v2f ec = *(const v2f*)&bf[4 * CHUNK + g0];
      v2f op = *(const v2f*)&bf[5 * CHUNK + g0];

      float dx0 = pxf - gx.x, dy0 = pyf - gy.x;
      float q0 = (ea.x * dx0) * dx0 + (eb.x * dx0) * dy0 + (ec.x * dy0) * dy0;
      float a0 = fminf(op.x * __builtin_amdgcn_exp2f(q0), MAX_ALPHA);
      float dx1 = pxf - gx.y, dy1 = pyf - gy.y;
      float q1 = (ea.y * dx1) * dx1 + (eb.y * dx1) * dy1 + (ec.y * dy1) * dy1;
      float a1 = fminf(op.y * __builtin_amdgcn_exp2f(q1), MAX_ALPHA);

      float m = (1.f - a0) * (1.f - a1);           // this half's factor
      float mo = __int_as_float(                    // other half's factor
          __builtin_amdgcn_ds_swizzle(__float_as_int(m), 0x401F)); // SWAPX16
      float pre = T * (h ? mo : 1.f);               // exclusive T at g0
      v2f A = {pre * a0, pre * (1.f - a0) * a1};
      T = T * m * mo;                               // identical in both halves

      v2f cb = *(const v2f*)&bf[ch6 * CHUNK + g0];
      v2f B = {cb.x * cmask, cb.y * cmask};

      acc = __builtin_amdgcn_wmma_f32_16x16x4_f32(
          false, A, false, B, (short)0, acc, false, false);
    }

    asm volatile("s_wait_asynccnt 0x0" ::: "memory");
    __syncthreads();
    cur ^= 1;
  }

  if (n16 < 3) {
#pragma unroll
    for (int r = 0; r < 8; ++r) {
      int p = pwave + 8 * h + r; // C layout: lane-half h holds rows M=8h..8h+7
      out[(cam * NPIX + p) * 3 + n16] = acc[r];
    }
  }
}

extern "C" void kernel_launch(void* const* d_in, const int* in_sizes, int n_in,
                              void* d_out, int out_size, void* d_ws, size_t ws_size,
                              hipStream_t stream) {
  (void)in_sizes; (void)n_in; (void)out_size; (void)ws_size;
  const float* vm    = (const float*)d_in[0];
  const float* Ks    = (const float*)d_in[1];
  const float* means = (const float*)d_in[2];
  const float* quats = (const float*)d_in[3];
  const float* logsc = (const float*)d_in[4];
  const float* opl   = (const float*)d_in[5];
  const float* cl    = (const float*)d_in[6];
  float* U = (float*)d_ws;            // 10 * FSTRIDE floats
  float* S = U + 10 * FSTRIDE;        //  9 * FSTRIDE floats
  float* out = (float*)d_out;

  prep_kernel<<<8, 256, 0, stream>>>(vm, Ks, means, quats, logsc, opl, cl, U);
  sort_kernel<<<2, 1024, 0, stream>>>(U, S);
  render_kernel<<<256, 256, 0, stream>>>(S, out);
}